// LinkPredictionModel_79714593014200
// MI455X (gfx1250) — compile-verified
//
#include <hip/hip_runtime.h>
#include <hip/hip_bf16.h>

// ---------------------------------------------------------------------------
// GCN link prediction for MI455X (gfx1250, wave32).
//   z1 = relu(GCNConv(x, W1, b1));  z2 = GCNConv(z1, W2, b2)
//   out[e] = dot(z2[a_e], z2[b_e])
// GEMMs use V_WMMA_F32_16X16X4_F32 (exact fp32 matrix path).
// Scatter uses hardware global_atomic_add_f32, wave-per-edge.
// ---------------------------------------------------------------------------

typedef __attribute__((ext_vector_type(2))) float v2f;
typedef __attribute__((ext_vector_type(8))) float v8f;

#define K_DIM 128   // both layers contract over 128 (D_IN = D_HID = 128)

__device__ __forceinline__ void atomic_add_f32(float* p, float v) {
  unsafeAtomicAdd(p, v);    // lowers to global_atomic_add_f32 on gfx1250
}

// ----------------------------- utility kernels -----------------------------

__global__ void zero_kernel(float* __restrict__ p, long long n) {
  long long i = (long long)blockIdx.x * blockDim.x + threadIdx.x;
  long long stride = (long long)gridDim.x * blockDim.x;
  for (; i < n; i += stride) p[i] = 0.0f;
}

__global__ void degree_kernel(const int* __restrict__ dst, float* __restrict__ deg, int ne) {
  int i = blockIdx.x * blockDim.x + threadIdx.x;
  int stride = gridDim.x * blockDim.x;
  for (; i < ne; i += stride) atomic_add_f32(&deg[dst[i]], 1.0f);
}

__global__ void dinv_kernel(float* __restrict__ deg, int n) {
  int i = blockIdx.x * blockDim.x + threadIdx.x;
  if (i < n) deg[i] = __frsqrt_rn(deg[i] + 1.0f);   // deg^{-1/2}, self loop included
}

// --------------------------- WMMA fp32 GEMM --------------------------------
// H[row0..row0+15, :] = X[row0..row0+15, 0..127] @ W[0..127, 0..DOUT-1]
// One block per 16-row tile. Wave w owns output columns [16w, 16w+16).
// A-layout (16x4 f32, 2 VGPRs): lanes 0-15 hold K={k,k+1} of row M=lane,
//   lanes 16-31 hold K={k+2,k+3} of row M=lane-16.
// B-layout mirrors A on the K split; C/D: VGPR j = row j (lanes 0-15) /
//   row j+8 (lanes 16-31), N = lane & 15.
template <int DOUT>
__global__ __launch_bounds__(32 * (DOUT / 16))
void gemm_wmma_kernel(const float* __restrict__ X, const float* __restrict__ W,
                      float* __restrict__ H) {
  constexpr int NW = DOUT / 16;
  __shared__ float xs[16 * K_DIM];            // 8 KB row-tile of X

  const int tid  = threadIdx.x;
  const int wave = tid >> 5;
  const int lane = tid & 31;
  const int row0 = blockIdx.x * 16;

  // cooperative, coalesced stage of the 16x128 X tile into LDS
  for (int i = tid; i < 16 * K_DIM; i += NW * 32)
    xs[i] = X[(long long)row0 * K_DIM + i];
  __syncthreads();

  const int mrow  = lane & 15;                // A row / B column within half-wave
  const int khalf = (lane >> 4) << 1;         // 0 for lanes 0-15, 2 for lanes 16-31
  const int n0    = wave * 16;                // this wave's output-column tile

  v8f acc = {0.f, 0.f, 0.f, 0.f, 0.f, 0.f, 0.f, 0.f};

#pragma unroll 4
  for (int k = 0; k < K_DIM; k += 4) {
    v2f a, b;
    a.x = xs[mrow * K_DIM + k + khalf];
    a.y = xs[mrow * K_DIM + k + khalf + 1];
    b.x = W[(k + khalf) * DOUT + n0 + mrow];
    b.y = W[(k + khalf + 1) * DOUT + n0 + mrow];
    // D = A(16x4) * B(4x16) + C   -> v_wmma_f32_16x16x4_f32
    acc = __builtin_amdgcn_wmma_f32_16x16x4_f32(
        /*neg_a=*/false, a, /*neg_b=*/false, b,
        /*c_mod=*/(short)0, acc, /*reuse_a=*/false, /*reuse_b=*/false);
  }

  const int mbase = (lane >> 4) << 3;         // 0 or 8
#pragma unroll
  for (int j = 0; j < 8; ++j)
    H[(long long)(row0 + mbase + j) * DOUT + n0 + mrow] = acc[j];
}

// ------------------------- edge scatter (wave/edge) ------------------------

template <int D>
__global__ void scatter_kernel(const float* __restrict__ h, const int* __restrict__ src,
                               const int* __restrict__ dst, const float* __restrict__ dinv,
                               float* __restrict__ agg, int ne) {
  const int e = blockIdx.x * (blockDim.x >> 5) + (threadIdx.x >> 5);
  if (e >= ne) return;
  const int lane = threadIdx.x & 31;
  const int s = src[e], d = dst[e];
  const float nrm = dinv[s] * dinv[d];
  const float* hs = h + (long long)s * D;
  float* ad = agg + (long long)d * D;
#pragma unroll
  for (int f = lane; f < D; f += 32)
    atomic_add_f32(&ad[f], hs[f] * nrm);
}

// self-loop term + bias (+ optional ReLU)
template <int D, bool RELU>
__global__ void finalize_kernel(float* __restrict__ agg, const float* __restrict__ h,
                                const float* __restrict__ dinv, const float* __restrict__ bias,
                                int n) {
  long long t = (long long)blockIdx.x * blockDim.x + threadIdx.x;
  if (t >= (long long)n * D) return;
  const int i = (int)(t / D);
  const int f = (int)(t % D);
  const float di = dinv[i];                   // dinv^2 == 1/deg
  float v = agg[t] + h[t] * di * di + bias[f];
  if (RELU) v = fmaxf(v, 0.0f);
  agg[t] = v;
}

// ------------------------------ decode -------------------------------------
// one wave per supervision edge; D_OUT = 64 -> 2 floats per lane
__global__ void decode_kernel(const float* __restrict__ z, const int* __restrict__ ea,
                              const int* __restrict__ eb, float* __restrict__ out, int nel) {
  const int e = blockIdx.x * (blockDim.x >> 5) + (threadIdx.x >> 5);
  if (e >= nel) return;
  const int lane = threadIdx.x & 31;
  const float* za = z + (long long)ea[e] * 64;
  const float* zb = z + (long long)eb[e] * 64;
  float s = za[lane] * zb[lane] + za[lane + 32] * zb[lane + 32];
#pragma unroll
  for (int off = 16; off > 0; off >>= 1)
    s += __shfl_xor(s, off, 32);
  if (lane == 0) out[e] = s;
}

// ------------------------------ launcher -----------------------------------

extern "C" void kernel_launch(void* const* d_in, const int* in_sizes, int n_in,
                              void* d_out, int out_size, void* d_ws, size_t ws_size,
                              hipStream_t stream) {
  const float* x   = (const float*)d_in[0];
  const int*   ei  = (const int*)  d_in[1];
  const int*   eli = (const int*)  d_in[2];
  const float* W1  = (const float*)d_in[3];
  const float* b1  = (const float*)d_in[4];
  const float* W2  = (const float*)d_in[5];
  const float* b2  = (const float*)d_in[6];
  float* out = (float*)d_out;

  const int N  = in_sizes[0] / 128;   // 100000 (divisible by 16)
  const int E  = in_sizes[1] / 2;     // 1.6M
  const int EL = in_sizes[2] / 2;     // 200k

  // workspace layout: h [N*128] | z [N*128] | dinv [N]
  float* h    = (float*)d_ws;
  float* z    = h + (long long)N * 128;
  float* dinv = z + (long long)N * 128;

  const int* src = ei;
  const int* dst = ei + E;
  const int* ea  = eli;
  const int* eb  = eli + EL;

  const int ZB = 2048;  // blocks for grid-stride zero/degree kernels

  // --- degrees: dinv = rsqrt(in_degree + 1)
  zero_kernel<<<ZB, 256, 0, stream>>>(dinv, N);
  degree_kernel<<<ZB, 256, 0, stream>>>(dst, dinv, E);
  dinv_kernel<<<(N + 255) / 256, 256, 0, stream>>>(dinv, N);

  // --- layer 1: h = x @ W1
  gemm_wmma_kernel<128><<<N / 16, 256, 0, stream>>>(x, W1, h);
  zero_kernel<<<ZB, 256, 0, stream>>>(z, (long long)N * 128);
  scatter_kernel<128><<<(E + 7) / 8, 256, 0, stream>>>(h, src, dst, dinv, z, E);
  finalize_kernel<128, true><<<(int)(((long long)N * 128 + 255) / 256), 256, 0, stream>>>(
      z, h, dinv, b1, N);

  // --- layer 2: h2 = z @ W2 (reuse h buffer: first N*64 floats)
  gemm_wmma_kernel<64><<<N / 16, 128, 0, stream>>>(z, W2, h);
  float* z2 = h + (long long)N * 64;          // free second half of h buffer
  zero_kernel<<<ZB, 256, 0, stream>>>(z2, (long long)N * 64);
  scatter_kernel<64><<<(E + 7) / 8, 256, 0, stream>>>(h, src, dst, dinv, z2, E);
  finalize_kernel<64, false><<<(int)(((long long)N * 64 + 255) / 256), 256, 0, stream>>>(
      z2, h, dinv, b2, N);

  // --- decode: out[e] = dot(z2[ea[e]], z2[eb[e]])
  decode_kernel<<<(EL + 7) / 8, 256, 0, stream>>>(z2, ea, eb, out, EL);
}